// local_gg_encoder_20495583936810
// MI455X (gfx1250) — compile-verified
//
#include <hip/hip_runtime.h>

// ---------------------------------------------------------------------------
// local_gg_encoder for MI455X (gfx1250, wave32)
//   h   = leaky(per-group dot(x, conv_w) + conv_b)          [128, 10000]
//   z   = h @ fc_w^T           (fc_b dropped: BN-invariant) [128, 2000]
//   out = leaky(BN_train(z) * gamma + beta)                 [128, 2000]
//
// HBM-bound problem (~190 MB traffic -> ~8us floor @ 23.3 TB/s), so:
//  - full fp32 path with V_WMMA_F32_16X16X4_F32 (no down-convert pass pays)
//  - GEMM stage double-buffers LDS via GLOBAL_LOAD_ASYNC_TO_LDS_B128
//    (ASYNCcnt) so the fc_w HBM stream overlaps the WMMA work
//  - BatchNorm fused into the GEMM epilogue (block owns the whole batch)
// ---------------------------------------------------------------------------

typedef __attribute__((ext_vector_type(2))) float v2f;
typedef __attribute__((ext_vector_type(8))) float v8f;

#define BATCH     128
#define NGROUP    10000
#define KWIN      20
#define LATENT    2000
#define KT        40            // K-chunk (10000 / 40 = 250 exact)
#define LDA       44            // padded LDS stride (conflict-free, 16B-aligned)
#define NCHUNK    (NGROUP / KT)
#define NEG_SLOPE 0.1f
#define BN_EPS    1e-5f

// ------------------------- Stage 1: grouped conv + leaky -------------------
__global__ __launch_bounds__(256) void conv_leaky_kernel(
    const float* __restrict__ x,       // [128, 200000]
    const float* __restrict__ cw,      // [10000, 20]
    const float* __restrict__ cb,      // [10000]
    float* __restrict__ h)             // [128, 10000]
{
    const int idx = blockIdx.x * 256 + threadIdx.x;      // exactly B*G threads
    const int b   = idx / NGROUP;
    const int g   = idx - b * NGROUP;
    // each group's 20 floats start at an 80-byte (16B-aligned) offset
    const float4* xg = (const float4*)(x + (size_t)b * (NGROUP * KWIN) + (size_t)g * KWIN);
    const float4* wg = (const float4*)(cw + (size_t)g * KWIN);
    float acc = cb[g];
#pragma unroll
    for (int i = 0; i < 5; ++i) {
        float4 xv = xg[i], wv = wg[i];
        acc += xv.x * wv.x + xv.y * wv.y + xv.z * wv.z + xv.w * wv.w;
    }
    acc = acc >= 0.f ? acc : NEG_SLOPE * acc;
    h[idx] = acc;
}

// ---- CDNA5 async global->LDS copy (ASYNCcnt-tracked, bypasses VGPRs) ------
__device__ __forceinline__ void async_b128(const float* gptr, const float* lptr) {
    unsigned lds = (unsigned)(unsigned long long)lptr;   // 32-bit LDS byte addr
    asm volatile("global_load_async_to_lds_b128 %0, %1, off"
                 :: "v"(lds), "v"(gptr) : "memory");
}
__device__ __forceinline__ void wait_async0() {
    asm volatile("s_wait_asynccnt 0x0" ::: "memory");
}

// --------------- Stage 2: WMMA GEMM + fused BatchNorm + leaky --------------
// One block per 16-column strip of the output. 8 waves, wave w owns rows
// [16w, 16w+16). The block sees the whole batch, so BN stats are block-local.
__global__ __launch_bounds__(256) void gemm_bn_leaky_kernel(
    const float* __restrict__ h,       // [128, 10000]
    const float* __restrict__ fw,      // [2000, 10000]
    const float* __restrict__ gamma,   // [2000]
    const float* __restrict__ beta,    // [2000]
    float* __restrict__ out)           // [128, 2000]
{
    __shared__ float As[2][BATCH * LDA];  // 2 x 128x40 (padded)  45056 B
    __shared__ float Bs[2][16 * LDA];     // 2 x  16x40 (padded)   5632 B
    __shared__ float redS[8][16];
    __shared__ float redQ[8][16];
    __shared__ float muS[16], rsS[16];

    const int tid  = threadIdx.x;
    const int wave = tid >> 5;
    const int lane = tid & 31;
    const int half = lane >> 4;        // K sub-pair select (ISA 16x4 f32 layout)
    const int mr   = lane & 15;        // row (A) / col (B) within tile
    const int c0   = blockIdx.x * 16;  // output column base

    // A tile: 128 rows x 10 float4 = 1280 loads -> 5/thread
    // B tile:  16 rows x 10 float4 =  160 loads -> threads 0..159
    const int arow0 = (tid + 0 * 256) / 10, af0 = (tid + 0 * 256) % 10;
    const int brow  = tid / 10,             bf   = tid % 10;

    auto issue_chunk = [&](int k0, int buf) {
#pragma unroll
        for (int i = 0; i < 5; ++i) {
            int li  = tid + i * 256;
            int row = li / 10;
            int f4  = li - row * 10;
            async_b128(h + (size_t)row * NGROUP + k0 + f4 * 4,
                       &As[buf][row * LDA + f4 * 4]);
        }
        if (tid < 160) {
            async_b128(fw + (size_t)(c0 + brow) * NGROUP + k0 + bf * 4,
                       &Bs[buf][brow * LDA + bf * 4]);
        }
    };

    v8f acc = {};                      // 16x16 f32 accumulator (8 VGPRs)

    // prologue: stage chunk 0
    issue_chunk(0, 0);
    wait_async0();
    __syncthreads();

    int p = 0;
    for (int c = 0; c < NCHUNK; ++c) {
        // overlap: fetch chunk c+1 into the back buffer while computing c
        if (c + 1 < NCHUNK) issue_chunk((c + 1) * KT, p ^ 1);

        // Fragments per ISA layout: lane<16 -> K={0,1}; lane>=16 -> K={2,3}
        const float* Arow = &As[p][(wave * 16 + mr) * LDA + 2 * half];
        const float* Brow = &Bs[p][mr * LDA + 2 * half];
#pragma unroll
        for (int s = 0; s < KT / 4; ++s) {
            v2f a = *(const v2f*)(Arow + 4 * s);
            v2f b = *(const v2f*)(Brow + 4 * s);
            acc = __builtin_amdgcn_wmma_f32_16x16x4_f32(
                /*neg_a=*/false, a, /*neg_b=*/false, b,
                /*c_mod=*/(short)0, acc, /*reuse_a=*/false, /*reuse_b=*/false);
        }

        wait_async0();      // chunk c+1 landed (overlapped with WMMAs above)
        __syncthreads();    // everyone done reading buf p / writing buf p^1
        p ^= 1;
    }

    // ---- fused BatchNorm (training-mode batch stats) + leaky ----
    // Lane L and L^16 hold the same column (c0+mr), rows r+{0,8} of the tile.
    float s = 0.f, q = 0.f;
#pragma unroll
    for (int r = 0; r < 8; ++r) { float v = acc[r]; s += v; q += v * v; }
    s += __shfl_xor(s, 16, 32);
    q += __shfl_xor(q, 16, 32);
    if (lane < 16) { redS[wave][lane] = s; redQ[wave][lane] = q; }
    __syncthreads();

    if (tid < 16) {
        float S = 0.f, Q = 0.f;
#pragma unroll
        for (int w = 0; w < 8; ++w) { S += redS[w][tid]; Q += redQ[w][tid]; }
        float mu  = S * (1.f / BATCH);
        float var = Q * (1.f / BATCH) - mu * mu;
        muS[tid] = mu;
        rsS[tid] = rsqrtf(var + BN_EPS);
    }
    __syncthreads();

    const float mu = muS[mr], rs = rsS[mr];
    const float ga = gamma[c0 + mr], be = beta[c0 + mr];
#pragma unroll
    for (int r = 0; r < 8; ++r) {
        int row = wave * 16 + r + 8 * half;
        float v = (acc[r] - mu) * rs * ga + be;
        v = v >= 0.f ? v : NEG_SLOPE * v;
        out[(size_t)row * LATENT + c0 + mr] = v;
    }
    (void)arow0; (void)af0;
}

// ---------------------------------------------------------------------------
extern "C" void kernel_launch(void* const* d_in, const int* in_sizes, int n_in,
                              void* d_out, int out_size, void* d_ws, size_t ws_size,
                              hipStream_t stream) {
    const float* x      = (const float*)d_in[0];
    const float* conv_w = (const float*)d_in[1];
    const float* conv_b = (const float*)d_in[2];
    const float* fc_w   = (const float*)d_in[3];
    // d_in[4] = fc_b: a per-column constant shift is exactly cancelled by the
    // training-mode BatchNorm mean subtraction (variance is shift-invariant
    // too), so it is intentionally unused.
    const float* gamma  = (const float*)d_in[5];
    const float* beta   = (const float*)d_in[6];
    float* out = (float*)d_out;
    float* h   = (float*)d_ws;                 // 128*10000*4 = 5.12 MB scratch

    conv_leaky_kernel<<<(BATCH * NGROUP) / 256, 256, 0, stream>>>(x, conv_w, conv_b, h);
    gemm_bn_leaky_kernel<<<LATENT / 16, 256, 0, stream>>>(h, fc_w, gamma, beta, out);

    (void)in_sizes; (void)n_in; (void)out_size; (void)ws_size;
}